// KGModel_56942676411131
// MI455X (gfx1250) — compile-verified
//
#include <hip/hip_runtime.h>
#include <hip/hip_bf16.h>

typedef __attribute__((ext_vector_type(2))) float v2f;
typedef __attribute__((ext_vector_type(8))) float v8f;

#define N_ENT   100000
#define DIM     256
#define BATCH   512
#define KDIM    512              // 2*DIM (real | imag concatenated)
#define NT_COLS 32               // entity columns per workgroup (2 WMMA n-tiles)
#define NEG_INF (-__builtin_inff())

// XOR bank swizzle: touches bits [5:2] only -> b64 pairs stay contiguous,
// 16 rows map to 16 distinct bank groups; lane halves (K vs K+2) differ in
// bit 1 so the full 32-lane ds read is bank-conflict-free with stride 512.
__device__ __forceinline__ int swz(int n) { return ((n & 7) << 3) | ((n & 8) >> 1); }

__device__ __forceinline__ unsigned ballot32(bool p) {
#if __has_builtin(__builtin_amdgcn_ballot_w32)
    return __builtin_amdgcn_ballot_w32(p);
#else
    return (unsigned)__ballot(p);
#endif
}

// ---------------------------------------------------------------------------
// Kernel 1: build A = [rr*sr - ri*si | rr*si + ri*sr] (512 x 512),
// precompute sc[b] = <A[b], B[correct_b]> and zero the rank counters.
// ---------------------------------------------------------------------------
__global__ __launch_bounds__(256) void prep_kernel(
    const float* __restrict__ node_r, const float* __restrict__ node_i,
    const float* __restrict__ rel_r,  const float* __restrict__ rel_i,
    const int*   __restrict__ triplets,
    float* __restrict__ A, float* __restrict__ sc, int* __restrict__ cnt)
{
    const int b = blockIdx.x;          // 0..511
    const int d = threadIdx.x;         // 0..255
    const int h = triplets[b * 3 + 0];
    const int r = triplets[b * 3 + 1];
    const int c = triplets[b * 3 + 2];
    const float sr = node_r[(size_t)h * DIM + d];
    const float si = node_i[(size_t)h * DIM + d];
    const float wr = rel_r[(size_t)r * DIM + d];
    const float wi = rel_i[(size_t)r * DIM + d];
    const float ar = wr * sr - wi * si;
    const float ai = wr * si + wi * sr;
    A[(size_t)b * KDIM + d]       = ar;
    A[(size_t)b * KDIM + DIM + d] = ai;

    // partial dot with the correct tail embedding
    __shared__ float red[256];
    red[d] = ar * node_r[(size_t)c * DIM + d] + ai * node_i[(size_t)c * DIM + d];
    __syncthreads();
    #pragma unroll
    for (int st = 128; st > 0; st >>= 1) {
        if (d < st) red[d] += red[d + st];
        __syncthreads();
    }
    if (d == 0) sc[b] = red[0];
    if (d < 3)  cnt[d * BATCH + b] = 0;   // zero counters every launch
}

// ---------------------------------------------------------------------------
// Kernel 2: scores = A(512xK) @ B(100000xK)^T via V_WMMA_F32_16X16X4_F32.
// One workgroup per 32-entity N-tile (3125 WGs). B-tile async-staged into
// 64 KB swizzled LDS. 8 waves x 4 M-tiles x 2 N-tiles cover 512x32.
// Epilogue writes all three score planes and ballots the rank counts.
// ---------------------------------------------------------------------------
__global__ __launch_bounds__(256) void gemm_kernel(
    const float* __restrict__ A,
    const float* __restrict__ node_r, const float* __restrict__ node_i,
    const unsigned char* __restrict__ tail_labels,
    const unsigned char* __restrict__ invalid_targets,
    const int*   __restrict__ triplets,
    const float* __restrict__ sc,
    int*   __restrict__ cnt,
    float* __restrict__ scores,
    float* __restrict__ filtered,
    float* __restrict__ target_filtered)
{
    __shared__ float Bt[NT_COLS * KDIM];     // exactly 64 KB, XOR-swizzled

    const int tid = threadIdx.x;
    const int n0  = blockIdx.x * NT_COLS;    // first entity column of this tile

    // ---- stage B tile: 32 rows x 512 k, 64 floats per thread -------------
    {
        const int n   = tid >> 3;            // 0..31 entity row within tile
        const int sub = tid & 7;             // 0..7  64-float chunk
        const int gn  = n0 + n;
        const int s   = swz(n);
        const float* src = (sub < 4)
            ? (node_r + (size_t)gn * DIM + sub * 64)
            : (node_i + (size_t)gn * DIM + (sub - 4) * 64);
        float* dstrow = &Bt[n * KDIM];
        const int kbase = sub * 64;
#if __has_builtin(__builtin_amdgcn_global_load_async_to_lds_b64)
        // builtin params are typed vector pointers: b64 -> v2i (AS1 src, AS3 dst)
        typedef int v2i_t __attribute__((vector_size(2 * sizeof(int))));
        typedef __attribute__((address_space(1))) v2i_t* g_v2i;
        typedef __attribute__((address_space(3))) v2i_t* l_v2i;
        #pragma unroll
        for (int p = 0; p < 32; ++p) {
            const int k = kbase + p * 2;
            __builtin_amdgcn_global_load_async_to_lds_b64(
                (g_v2i)(src + p * 2),
                (l_v2i)(dstrow + (k ^ s)), 0, 0);
        }
#if __has_builtin(__builtin_amdgcn_s_wait_asynccnt)
        __builtin_amdgcn_s_wait_asynccnt(0);
#else
        asm volatile("s_wait_asynccnt 0x0" ::: "memory");
#endif
#else
        #pragma unroll
        for (int p = 0; p < 32; ++p) {
            const int k = kbase + p * 2;
            const float2 v = *(const float2*)(src + p * 2);
            dstrow[(k ^ s) + 0] = v.x;
            dstrow[(k ^ s) + 1] = v.y;
        }
#endif
    }
    __syncthreads();

    const int wv   = tid >> 5;               // wave id 0..7 (wave32)
    const int lane = tid & 31;
    const int row  = lane & 15;              // fragment row (A) / col (B)
    const int koff = (lane >> 4) << 1;       // lanes 0-15: K+{0,1}; 16-31: K+{2,3}
    const int sw   = swz(row);

    const float* aBase[4];
    #pragma unroll
    for (int t = 0; t < 4; ++t)
        aBase[t] = A + (size_t)((wv + t * 8) * 16 + row) * KDIM + koff;

    v8f acc[4][2];
    #pragma unroll
    for (int t = 0; t < 4; ++t) {
        acc[t][0] = (v8f){0.f,0.f,0.f,0.f,0.f,0.f,0.f,0.f};
        acc[t][1] = (v8f){0.f,0.f,0.f,0.f,0.f,0.f,0.f,0.f};
    }

    for (int k0 = 0; k0 < KDIM; k0 += 4) {
        const int kk = (k0 + koff) ^ sw;
        const v2f bf0 = *(const v2f*)&Bt[ row       * KDIM + kk];
        const v2f bf1 = *(const v2f*)&Bt[(row + 16) * KDIM + kk];
        #pragma unroll
        for (int t = 0; t < 4; ++t) {
            const v2f af = *(const v2f*)(aBase[t] + k0);   // reused for 2 n-tiles
            acc[t][0] = __builtin_amdgcn_wmma_f32_16x16x4_f32(
                false, af, false, bf0, (short)0, acc[t][0], false, false);
            acc[t][1] = __builtin_amdgcn_wmma_f32_16x16x4_f32(
                false, af, false, bf1, (short)0, acc[t][1], false, false);
        }
    }

    // ---- epilogue: C/D layout = VGPR j : lanes 0-15 M=j, lanes 16-31 M=j+8 --
    const int col   = lane & 15;
    const int hi8   = (lane >> 4) << 3;
    const int gcol0 = n0 + col;
    const int gcol1 = n0 + 16 + col;
    const bool inv0 = invalid_targets[gcol0] != 0;
    const bool inv1 = invalid_targets[gcol1] != 0;

    #pragma unroll
    for (int t = 0; t < 4; ++t) {
        const int m16 = (wv + t * 8) * 16;
        #pragma unroll
        for (int j = 0; j < 8; ++j) {
            const int brow = m16 + j + hi8;
            const int corr = triplets[brow * 3 + 2];
            const float scv = sc[brow];

            const size_t off0 = (size_t)brow * N_ENT + gcol0;
            const size_t off1 = (size_t)brow * N_ENT + gcol1;
            const float s0 = acc[t][0][j];
            const float s1 = acc[t][1][j];
            const bool tl0 = tail_labels[off0] != 0;
            const bool tl1 = tail_labels[off1] != 0;
            const float f0 = tl0 ? NEG_INF : s0;
            const float f1 = tl1 ? NEG_INF : s1;
            const float g0 = inv0 ? NEG_INF : f0;
            const float g1 = inv1 ? NEG_INF : f1;
            scores[off0] = s0;           scores[off1] = s1;
            filtered[off0] = f0;         filtered[off1] = f1;
            target_filtered[off0] = g0;  target_filtered[off1] = g1;

            // rank predicates (exclude the correct column itself: its WMMA
            // value may differ from the scalar sc by an ulp)
            const bool cs0 = (gcol0 != corr) && ((s0 > scv) || (s0 == scv && gcol0 < corr));
            const bool cs1 = (gcol1 != corr) && ((s1 > scv) || (s1 == scv && gcol1 < corr));
            const bool cf0 = cs0 && !tl0;
            const bool cf1 = cs1 && !tl1;
            const bool cg0 = cf0 && !inv0;
            const bool cg1 = cf1 && !inv1;

            const unsigned ms0 = ballot32(cs0), ms1 = ballot32(cs1);
            const unsigned mf0 = ballot32(cf0), mf1 = ballot32(cf1);
            const unsigned mg0 = ballot32(cg0), mg1 = ballot32(cg1);

            if (lane == 0) {
                const int rA = m16 + j;       // rows held by lanes 0-15
                const int rB = m16 + 8 + j;   // rows held by lanes 16-31
                atomicAdd(&cnt[0 * BATCH + rA], __popc(ms0 & 0xffffu) + __popc(ms1 & 0xffffu));
                atomicAdd(&cnt[0 * BATCH + rB], __popc(ms0 >> 16)     + __popc(ms1 >> 16));
                atomicAdd(&cnt[1 * BATCH + rA], __popc(mf0 & 0xffffu) + __popc(mf1 & 0xffffu));
                atomicAdd(&cnt[1 * BATCH + rB], __popc(mf0 >> 16)     + __popc(mf1 >> 16));
                atomicAdd(&cnt[2 * BATCH + rA], __popc(mg0 & 0xffffu) + __popc(mg1 & 0xffffu));
                atomicAdd(&cnt[2 * BATCH + rB], __popc(mg0 >> 16)     + __popc(mg1 >> 16));
            }
        }
    }
}

// ---------------------------------------------------------------------------
// Kernel 3: rank = count + 1 (1536 values)
// ---------------------------------------------------------------------------
__global__ __launch_bounds__(256) void finish_kernel(
    const int* __restrict__ cnt, int* __restrict__ ranks)
{
    const int i = blockIdx.x * 256 + threadIdx.x;
    if (i < 3 * BATCH) ranks[i] = cnt[i] + 1;
}

// ---------------------------------------------------------------------------
extern "C" void kernel_launch(void* const* d_in, const int* in_sizes, int n_in,
                              void* d_out, int out_size, void* d_ws, size_t ws_size,
                              hipStream_t stream)
{
    (void)in_sizes; (void)n_in; (void)out_size; (void)ws_size;

    const float* node_r = (const float*)d_in[0];
    const float* node_i = (const float*)d_in[1];
    const float* rel_r  = (const float*)d_in[2];
    const float* rel_i  = (const float*)d_in[3];
    const int*   triplets = (const int*)d_in[4];
    const unsigned char* tail_labels     = (const unsigned char*)d_in[5];
    const unsigned char* invalid_targets = (const unsigned char*)d_in[6];

    float* out = (float*)d_out;
    const size_t PLANE = (size_t)BATCH * N_ENT;
    float* scores          = out;
    float* filtered        = out + PLANE;
    float* target_filtered = out + 2 * PLANE;
    int*   ranks           = (int*)(out + 3 * PLANE);   // 3x512 int32, in order

    float* A  = (float*)d_ws;                 // 512 x 512 f32 = 1 MB
    float* sc = A + (size_t)BATCH * KDIM;     // 512 f32
    int*   cnt = (int*)(sc + BATCH);          // 3 x 512 int32

    prep_kernel<<<BATCH, 256, 0, stream>>>(node_r, node_i, rel_r, rel_i,
                                           triplets, A, sc, cnt);

    gemm_kernel<<<N_ENT / NT_COLS, 256, 0, stream>>>(
        A, node_r, node_i, tail_labels, invalid_targets, triplets, sc, cnt,
        scores, filtered, target_filtered);

    finish_kernel<<<6, 256, 0, stream>>>(cnt, ranks);
}